// GNN_50019189129858
// MI455X (gfx1250) — compile-verified
//
#include <hip/hip_runtime.h>
#include <hip/hip_bf16.h>

// GCN pipeline for MI455X (gfx1250), wave32.
// CSR build (count/scan/fill) once -> [gemm(WMMA f32) -> gather/combine] x3 -> fused pool+proj.
// No f32 atomics in the hot path; all feature traffic is streaming + L2-resident.

typedef __attribute__((ext_vector_type(2))) float v2f;
typedef __attribute__((ext_vector_type(8))) float v8f;

#define HDIM 128

// ---------------- int zero ----------------
__global__ void zero_i_kernel(int* __restrict__ p, int n) {
  int i = blockIdx.x * blockDim.x + threadIdx.x;
  if (i < n) p[i] = 0;
}

// ---------------- in-degree counts ----------------
__global__ void count_kernel(const int* __restrict__ dst, int* __restrict__ cnt, int e) {
  int i = blockIdx.x * blockDim.x + threadIdx.x;
  if (i < e) atomicAdd(&cnt[dst[i]], 1);
}

// dinv_sqrt[i] = rsqrt(deg_i + 1)  (self-loop included)
__global__ void dinv_kernel(const int* __restrict__ cnt, float* __restrict__ dinv, int n) {
  int i = blockIdx.x * blockDim.x + threadIdx.x;
  if (i < n) dinv[i] = rsqrtf((float)cnt[i] + 1.0f);
}

// ---------------- two-level exclusive scan over counts -> rowptr ----------------
__global__ __launch_bounds__(256) void scan_block_kernel(const int* __restrict__ cnt, int n,
                                                         int* __restrict__ rowptr,
                                                         int* __restrict__ blockSums) {
  __shared__ int sh[256];
  const int tid = threadIdx.x;
  const int gid = blockIdx.x * 256 + tid;
  int v = (gid < n) ? cnt[gid] : 0;
  sh[tid] = v;
  __syncthreads();
  for (int off = 1; off < 256; off <<= 1) {          // inclusive Hillis-Steele
    int t = (tid >= off) ? sh[tid - off] : 0;
    __syncthreads();
    sh[tid] += t;
    __syncthreads();
  }
  if (gid < n) rowptr[gid] = sh[tid] - v;            // exclusive
  if (tid == 255) blockSums[blockIdx.x] = sh[255];
}

// single-block exclusive scan of block sums (nb <= 256)
__global__ __launch_bounds__(256) void scan_top_kernel(int* __restrict__ blockSums, int nb) {
  __shared__ int sh[256];
  const int tid = threadIdx.x;
  int v = (tid < nb) ? blockSums[tid] : 0;
  sh[tid] = v;
  __syncthreads();
  for (int off = 1; off < 256; off <<= 1) {
    int t = (tid >= off) ? sh[tid - off] : 0;
    __syncthreads();
    sh[tid] += t;
    __syncthreads();
  }
  if (tid < nb) blockSums[tid] = sh[tid] - v;        // exclusive offsets
}

// rowptr += block offset ; also seed cursor
__global__ void add_offsets_kernel(int* __restrict__ rowptr, const int* __restrict__ blockSums,
                                   int* __restrict__ cursor, int n) {
  int i = blockIdx.x * blockDim.x + threadIdx.x;
  if (i < n) {
    int r = rowptr[i] + blockSums[i >> 8];
    rowptr[i] = r;
    cursor[i] = r;
  }
}

// scatter edge ids into CSR slots
__global__ void fill_kernel(const int* __restrict__ dst, int* __restrict__ cursor,
                            int* __restrict__ eidx, int e) {
  int i = blockIdx.x * blockDim.x + threadIdx.x;
  if (i < e) {
    int pos = atomicAdd(&cursor[dst[i]], 1);
    eidx[pos] = i;
  }
}

// ---------------- GEMM: out[n][128] = h[n][128] @ W[128][128] via V_WMMA_F32_16X16X4_F32 ----
// 256 threads = 8 waves; each wave computes a 16-row x 128-col strip; block covers 128 rows.
// W packed in LDS as float4 per (k-group-of-4, n): dword ((k>>2)*128+n)*4 + (k&3).
// -> each B fragment is one aligned ds_load_b64 straight into the WMMA operand pair,
//    and a tile's 32 lanes cover 64 consecutive dwords = all 64 banks (conflict-free).
__global__ __launch_bounds__(256) void gemm_wmma_kernel(const float* __restrict__ h,
                                                        const float* __restrict__ W,
                                                        float* __restrict__ out,
                                                        int nRows) {
  __shared__ float lW[HDIM * HDIM]; // 64KB packed
  {
    const float4* w4 = (const float4*)W;
    for (int i = threadIdx.x; i < (HDIM * HDIM) / 4; i += 256) {
      float4 v = w4[i];
      int k  = i >> 5;           // (4*i) >> 7
      int n0 = (4 * i) & 127;
      int base = ((k >> 2) * HDIM + n0) * 4 + (k & 3);
      lW[base]      = v.x;
      lW[base + 4]  = v.y;
      lW[base + 8]  = v.z;
      lW[base + 12] = v.w;
    }
  }
  __syncthreads();

  const int lane    = threadIdx.x & 31;
  const int wave    = threadIdx.x >> 5;
  const int rowBase = blockIdx.x * 128 + wave * 16;
  const int mr      = lane & 15;        // row-in-tile (A) / col-in-tile (B,D)
  const int khalf   = (lane >> 4) * 2;  // lanes 0-15: K0,K1 ; lanes 16-31: K2,K3

  int m = rowBase + mr;
  if (m >= nRows) m = nRows - 1;        // clamp loads; stores masked below
  const float2* arow = (const float2*)(h + (size_t)m * HDIM);
  const float2* lp   = (const float2*)lW;

  const v8f zero8 = {0.f, 0.f, 0.f, 0.f, 0.f, 0.f, 0.f, 0.f};
  v8f acc[8];
#pragma unroll
  for (int t = 0; t < 8; ++t) acc[t] = zero8;

  for (int k0 = 0; k0 < HDIM; k0 += 4) {
    float2 af = arow[(k0 + khalf) >> 1];
    v2f a; a.x = af.x; a.y = af.y;
#pragma unroll
    for (int nt = 0; nt < 8; ++nt) {
      const int n = nt * 16 + mr;
      float2 bf = lp[((k0 >> 2) * HDIM + n) * 2 + (khalf >> 1)];
      v2f b; b.x = bf.x; b.y = bf.y;
      acc[nt] = __builtin_amdgcn_wmma_f32_16x16x4_f32(
          /*neg_a=*/false, a, /*neg_b=*/false, b,
          /*c_mod=*/(short)0, acc[nt], /*reuse_a=*/false, /*reuse_b=*/false);
    }
  }

  // D layout: col = lane&15 ; rows = rowBase + 8*(lane>=16) + vgpr
  const int rowOff = (lane >> 4) * 8;
#pragma unroll
  for (int nt = 0; nt < 8; ++nt) {
#pragma unroll
    for (int r = 0; r < 8; ++r) {
      const int row = rowBase + rowOff + r;
      if (row < nRows) out[(size_t)row * HDIM + nt * 16 + mr] = acc[nt][r];
    }
  }
}

// ---------------- fused neighbor gather + self-loop + bias (+relu) ----------------
// one 128-thread block per destination node; edge list staged through LDS in chunks.
// hout[d][f] = relu( dinv[d]*sum_e hw[src_e][f]*dinv[src_e] + dinv[d]^2*hw[d][f] + bias[f] )
__global__ __launch_bounds__(128) void gather_combine_kernel(
    const float* __restrict__ hw, const int* __restrict__ src,
    const int* __restrict__ rowptr, const int* __restrict__ cnt,
    const int* __restrict__ eidx, const float* __restrict__ dinv,
    const float* __restrict__ bias, float* __restrict__ hout, int relu) {
  __shared__ int   sS[128];
  __shared__ float sW[128];
  const int d = blockIdx.x;
  const int f = threadIdx.x;
  const int beg = rowptr[d];
  const int end = beg + cnt[d];
  const float dd = dinv[d];

  float sum = 0.0f;
  for (int base = beg; base < end; base += 128) {
    const int j = base + f;
    if (j < end) {
      int s = src[eidx[j]];
      sS[f] = s;
      sW[f] = dinv[s];
    }
    __syncthreads();
    const int c = (end - base < 128) ? (end - base) : 128;
    for (int t = 0; t < c; ++t)
      sum += hw[(size_t)sS[t] * HDIM + f] * sW[t];
    __syncthreads();
  }

  float v = sum * dd + hw[(size_t)d * HDIM + f] * dd * dd + bias[f];
  if (relu) v = fmaxf(v, 0.0f);
  hout[(size_t)d * HDIM + f] = v;
}

// ---------------- fused global_add_pool + output projection ----------------
// batch is sorted: block g binary-searches its node range, accumulates per-feature,
// dots with Wo, tree-reduces in LDS. Fully deterministic, no atomics.
__global__ __launch_bounds__(128) void pool_proj_kernel(const float* __restrict__ h,
                                                        const int* __restrict__ batch, int n,
                                                        const float* __restrict__ Wo,
                                                        const float* __restrict__ bo,
                                                        float* __restrict__ out) {
  const int g = blockIdx.x;
  const int f = threadIdx.x;
  int lo = 0, hi = n;
  while (lo < hi) { int mid = (lo + hi) >> 1; if (batch[mid] < g) lo = mid + 1; else hi = mid; }
  const int start = lo;
  hi = n;
  while (lo < hi) { int mid = (lo + hi) >> 1; if (batch[mid] < g + 1) lo = mid + 1; else hi = mid; }
  const int stop = lo;

  float s = 0.0f;
  for (int i = start; i < stop; ++i) s += h[(size_t)i * HDIM + f];

  __shared__ float red[HDIM];
  red[f] = s * Wo[f];
  __syncthreads();
#pragma unroll
  for (int st = 64; st > 0; st >>= 1) {
    if (f < st) red[f] += red[f + st];
    __syncthreads();
  }
  if (f == 0) out[g] = red[0] + bo[0];
}

extern "C" void kernel_launch(void* const* d_in, const int* in_sizes, int n_in,
                              void* d_out, int out_size, void* d_ws, size_t ws_size,
                              hipStream_t stream) {
  const float* x   = (const float*)d_in[0];   // [N,128]
  const int*   ei  = (const int*)d_in[1];     // [2,E] row0=src row1=dst
  const int*   bat = (const int*)d_in[2];     // [N] sorted
  const float* W1  = (const float*)d_in[3];
  const float* b1  = (const float*)d_in[4];
  const float* W2  = (const float*)d_in[5];
  const float* b2  = (const float*)d_in[6];
  const float* W3  = (const float*)d_in[7];
  const float* b3  = (const float*)d_in[8];
  const float* Wo  = (const float*)d_in[9];
  const float* bo  = (const float*)d_in[10];
  float*       out = (float*)d_out;

  const int N = in_sizes[0] / HDIM;   // 50000
  const int E = in_sizes[1] / 2;      // 800000
  const int G = out_size;             // 512
  const long NH = (long)N * HDIM;
  const int NPAD = 51200;             // padded node stride for ws carving

  float* ws   = (float*)d_ws;
  float* dinv = ws;                   // NPAD floats
  float* hw   = ws + NPAD;            // N*H
  float* hbuf = hw + NH;              // N*H
  int* ib        = (int*)(hbuf + NH);
  int* cnt       = ib;                // NPAD
  int* rowptr    = cnt + NPAD;        // NPAD
  int* cursor    = rowptr + NPAD;     // NPAD
  int* blockSums = cursor + NPAD;     // 256
  int* eidx      = blockSums + 256;   // E

  const int* src = ei;
  const int* dst = ei + E;

  const int TB = 256;
  auto blocks = [&](long n) { return (unsigned)((n + TB - 1) / TB); };
  const unsigned gemmBlocks = (unsigned)((N + 127) / 128);
  const int nb = (N + 255) / 256;     // 196 (<=256 required by scan_top)

  // ---- CSR build + normalization (graph fixed across all 3 layers) ----
  zero_i_kernel<<<blocks(N), TB, 0, stream>>>(cnt, N);
  count_kernel<<<blocks(E), TB, 0, stream>>>(dst, cnt, E);
  dinv_kernel<<<blocks(N), TB, 0, stream>>>(cnt, dinv, N);
  scan_block_kernel<<<(unsigned)nb, 256, 0, stream>>>(cnt, N, rowptr, blockSums);
  scan_top_kernel<<<1, 256, 0, stream>>>(blockSums, nb);
  add_offsets_kernel<<<blocks(N), TB, 0, stream>>>(rowptr, blockSums, cursor, N);
  fill_kernel<<<blocks(E), TB, 0, stream>>>(dst, cursor, eidx, E);

  // ---- layer 1 (relu): x -> hbuf ----
  gemm_wmma_kernel<<<gemmBlocks, TB, 0, stream>>>(x, W1, hw, N);
  gather_combine_kernel<<<(unsigned)N, HDIM, 0, stream>>>(hw, src, rowptr, cnt, eidx, dinv, b1, hbuf, 1);

  // ---- layer 2 (relu): hbuf -> hbuf ----
  gemm_wmma_kernel<<<gemmBlocks, TB, 0, stream>>>(hbuf, W2, hw, N);
  gather_combine_kernel<<<(unsigned)N, HDIM, 0, stream>>>(hw, src, rowptr, cnt, eidx, dinv, b2, hbuf, 1);

  // ---- layer 3 (no relu): hbuf -> hbuf ----
  gemm_wmma_kernel<<<gemmBlocks, TB, 0, stream>>>(hbuf, W3, hw, N);
  gather_combine_kernel<<<(unsigned)N, HDIM, 0, stream>>>(hw, src, rowptr, cnt, eidx, dinv, b3, hbuf, 0);

  // ---- fused pool + projection ----
  pool_proj_kernel<<<(unsigned)G, HDIM, 0, stream>>>(hbuf, bat, N, Wo, bo, out);
}